// SiegelmannSontag1_80857054315132
// MI455X (gfx1250) — compile-verified
//
#include <hip/hip_runtime.h>

// ---------------------------------------------------------------------------
// SiegelmannSontag1 (S=4, P=6) for MI455X / gfx1250
// Dominant cost: streaming cd_w (442MB) + gamma_w (204MB) + beta_w (34MB).
// Strategy: one-pass grid-stride streaming reduction built on
// V_WMMA_F32_16X16X4_F32 (f32 in/out, no precision loss), deterministic
// two-stage reduction (per-block partials in d_ws -> single-wave finisher).
// ---------------------------------------------------------------------------

namespace {
constexpr int  SDIM      = 4;
constexpr int  PDIM      = 6;
constexpr int  KIN       = 8 * PDIM + SDIM;   // 52 input features to cd
constexpr int  NCOMP     = SDIM + 4 * PDIM;   // 28 streamed output components
constexpr long ROWS      = 2125764;           // 4 * 9^6
constexpr long TILES     = ROWS / 16;         // 132860 full 16-row tiles
constexpr long REM_START = TILES * 16;        // 4 remainder rows handled in k2
constexpr int  NTHR      = 256;               // 8 waves / block
constexpr int  WPB       = NTHR / 32;
constexpr int  NBLK      = 512;               // needs NBLK*NCOMP*4 = 56KB ws
}

typedef __attribute__((ext_vector_type(2))) float v2f;
typedef __attribute__((ext_vector_type(8))) float v8f;

__device__ __forceinline__ float swap_halves(float x) {
  // ds_swizzle SWAPX16: and=0x1f, or=0x00, xor=0x10 -> imm 0x401F
  return __int_as_float(__builtin_amdgcn_ds_swizzle(__float_as_int(x), 0x401F));
}
__device__ __forceinline__ float satf(float v) {
  return fminf(fmaxf(v, 0.0f), 1.0f);
}

// ---------------------------------------------------------------------------
// Kernel 1: stream rows [0, TILES*16), accumulate 28 component partial sums
// per block into ws[block*NCOMP + c].
// ---------------------------------------------------------------------------
__global__ __launch_bounds__(NTHR)
void ss1_stream_k1(const float* __restrict__ x,
                   const float* __restrict__ cd_w,
                   const float* __restrict__ cd_b,
                   const float* __restrict__ beta_w,
                   const float* __restrict__ gamma_w,
                   float* __restrict__ ws)
{
  const int lane = threadIdx.x & 31;
  const int wib  = threadIdx.x >> 5;                 // wave in block
  const int lh   = lane >> 4;                        // half-wave: 0|1
  const int lm   = lane & 15;
  const long gwave  = (long)blockIdx.x * WPB + wib;
  const long nwaves = (long)gridDim.x * WPB;

  // --- loop-invariant B chunks: w_in replicated into all 16 columns ---
  // w_in = [x[28:52] (top), x[52:76] (nonempty), x[0:4] (state)]
  // B layout (4x16 f32): vgpr0 = {K0 | K2}, vgpr1 = {K1 | K3} across halves.
  v2f bch[13];
#pragma unroll
  for (int c = 0; c < 13; ++c) {
    const float* p = (c < 12) ? (x + 28 + 4 * c) : x;   // chunk 12 = state
    v2f b;
    b.x = p[2 * lh];
    b.y = p[2 * lh + 1];
    bch[c] = b;
  }

  // --- per-lane row pointers for the stage-2 A tiles (28 comps + 4 pad) ---
  const float* rowp[2];
#pragma unroll
  for (int t = 0; t < 2; ++t) {
    int m = 16 * t + lm;                     // component index 0..31
    if (m < SDIM) {
      rowp[t] = beta_w + (long)m * ROWS;
    } else {
      int g = m - SDIM;                      // gamma row
      if (g > 4 * PDIM - 1) g = 4 * PDIM - 1;   // clamp pad rows (ignored later)
      rowp[t] = gamma_w + (long)g * ROWS;
    }
  }

  v8f C0 = {0.f,0.f,0.f,0.f,0.f,0.f,0.f,0.f};   // comps 0..15
  v8f C1 = {0.f,0.f,0.f,0.f,0.f,0.f,0.f,0.f};   // comps 16..27 (+pad)

  for (long tile = gwave; tile < TILES; tile += nwaves) {
    const long r = tile * 16;

    // uniform-branch prefetch of next tile's cd_w rows (EXEC stays all-1s)
    if (tile + nwaves < TILES) {
      const long rn = (tile + nwaves) * 16;
      __builtin_prefetch(cd_w + (rn + lm) * KIN, 0, 0);
    }

    // --- init accumulator D with bias cd_b[r+M] via readlane broadcast ---
    const int tb = __float_as_int(cd_b[r + lm]);   // lane lm holds cd_b[r+lm]
    v8f D;
#pragma unroll
    for (int v = 0; v < 8; ++v) {
      float blo = __int_as_float(__builtin_amdgcn_readlane(tb, v));
      float bhi = __int_as_float(__builtin_amdgcn_readlane(tb, v + 8));
      D[v] = lh ? bhi : blo;
    }

    // --- stage 1: D += cd_w[r:r+16, :] x w_in  (13 WMMAs, K=4 each) ---
    const float* arow = cd_w + (r + lm) * KIN;
#pragma unroll
    for (int c = 0; c < 13; ++c) {
      v2f a = *(const v2f*)(arow + 4 * c + 2 * lh);
      D = __builtin_amdgcn_wmma_f32_16x16x4_f32(false, a, false, bch[c],
                                                (short)0, D, false, false);
    }

    // --- saturate: a-values, replicated across all 16 columns ---
    v8f A8;
#pragma unroll
    for (int v = 0; v < 8; ++v) A8[v] = satf(D[v]);

    // --- stage 2: C += [beta;gamma][:, r:r+16] x a  (8 WMMAs) ---
#pragma unroll
    for (int c2 = 0; c2 < 4; ++c2) {
      v2f B2;
      if (c2 < 2) {
        // rows a[4c2 .. 4c2+3] live in low halves of vgprs 4c2..4c2+3
        const int v0 = 4 * c2;
        float s2 = swap_halves(A8[v0 + 2]);
        float s3 = swap_halves(A8[v0 + 3]);
        B2.x = lh ? s2 : A8[v0];
        B2.y = lh ? s3 : A8[v0 + 1];
      } else {
        // rows a[8+.. ] live in high halves of vgprs 0..7
        const int v0 = 4 * (c2 - 2);
        float s0 = swap_halves(A8[v0]);
        float s1 = swap_halves(A8[v0 + 1]);
        B2.x = lh ? A8[v0 + 2] : s0;
        B2.y = lh ? A8[v0 + 3] : s1;
      }
      const long col = r + 4 * c2 + 2 * lh;
      v2f A20 = *(const v2f*)(rowp[0] + col);
      v2f A21 = *(const v2f*)(rowp[1] + col);
      C0 = __builtin_amdgcn_wmma_f32_16x16x4_f32(false, A20, false, B2,
                                                 (short)0, C0, false, false);
      C1 = __builtin_amdgcn_wmma_f32_16x16x4_f32(false, A21, false, B2,
                                                 (short)0, C1, false, false);
    }
  }

  // --- deterministic block reduction: wave partials -> LDS -> ws ---
  __shared__ float lds[WPB * 32];
  if (lm == 0) {                   // lanes 0 and 16 hold column 0 of C tiles
#pragma unroll
    for (int v = 0; v < 8; ++v) {
      const int c0 = v + 8 * lh;                 // 0..15, all valid
      lds[wib * 32 + c0] = C0[v];
      const int c1 = 16 + v + 8 * lh;            // 16..31; guard pad
      if (c1 < NCOMP) lds[wib * 32 + c1] = C1[v];
    }
  }
  __syncthreads();
  if (threadIdx.x < NCOMP) {
    float s = 0.0f;
#pragma unroll
    for (int w = 0; w < WPB; ++w) s += lds[w * 32 + threadIdx.x];
    ws[(long)blockIdx.x * NCOMP + threadIdx.x] = s;
  }
}

// ---------------------------------------------------------------------------
// Kernel 2: single wave. Sum block partials in fixed order, add the 4
// remainder rows scalarly, then do all the tiny 24/6-dim linear algebra.
// ---------------------------------------------------------------------------
__global__ __launch_bounds__(32)
void ss1_finish_k2(const float* __restrict__ x,
                   const float* __restrict__ cd_w,
                   const float* __restrict__ cd_b,
                   const float* __restrict__ beta_w,
                   const float* __restrict__ gamma_w,
                   const float* __restrict__ w_stack,
                   const float* __restrict__ w_nss,
                   const float* __restrict__ b_nss,
                   const float* __restrict__ w_nst,
                   const float* __restrict__ w_st,
                   const float* __restrict__ b_st,
                   const float* __restrict__ w_sn,
                   const float* __restrict__ b_sn,
                   const float* __restrict__ ws,
                   int nblk,
                   float* __restrict__ out)
{
  const int j = threadIdx.x;
  __shared__ float sh[64];

  if (j < NCOMP) {
    float s = 0.0f;
    for (int b = 0; b < nblk; ++b) s += ws[(long)b * NCOMP + j];

    // remainder rows REM_START .. ROWS-1 (4 rows), scalar
    float wv[KIN];
#pragma unroll
    for (int c = 0; c < KIN; ++c) wv[c] = (c < 48) ? x[28 + c] : x[c - 48];
    const float* mp = (j < SDIM) ? (beta_w + (long)j * ROWS)
                                 : (gamma_w + (long)(j - SDIM) * ROWS);
    for (long row = REM_START; row < ROWS; ++row) {
      const float* wr = cd_w + row * KIN;
      float acc = cd_b[row];
#pragma unroll
      for (int c = 0; c < KIN; ++c) acc += wr[c] * wv[c];
      s += mp[row] * satf(acc);
    }
    sh[j] = s;
  }
  __syncthreads();

  if (j < SDIM) out[j] = sh[j];                       // next_state

  if (j < 4 * PDIM) {
    // nnss[j] = w_nss@stack + b_nss + w_nst@top + gamma_sum - 1
    float nn = b_nss[j] + sh[SDIM + j] - 1.0f;
#pragma unroll
    for (int k = 0; k < PDIM; ++k) {
      float sk = 0.0f, tk = 0.0f;
      for (int i = 0; i < 4 * PDIM; ++i) {
        const float wsk = w_stack[k * (4 * PDIM) + i];
        sk += wsk * satf(x[SDIM + i]);                // stack = w_stack@sat(nss)
        tk += wsk * satf(x[SDIM + 4 * PDIM + i]);     // top   = w_stack@sat(nst)
      }
      nn += w_nss[j * PDIM + k] * sk + w_nst[j * PDIM + k] * tk;
    }
    out[SDIM + j] = nn;
    sh[32 + j] = nn;
  }
  __syncthreads();

  if (j < 4 * PDIM) {
    float t1 = b_st[j], t2 = b_sn[j];
    for (int i = 0; i < 4 * PDIM; ++i) {
      const float nv = sh[32 + i];
      t1 += w_st[j * (4 * PDIM) + i] * nv;
      t2 += w_sn[j * (4 * PDIM) + i] * nv;
    }
    out[SDIM + 4 * PDIM + j]  = t1;                   // next_noisy_sub_top
    out[SDIM + 8 * PDIM + j]  = t2;                   // next_noisy_sub_nonempty
  }
}

// ---------------------------------------------------------------------------
extern "C" void kernel_launch(void* const* d_in, const int* in_sizes, int n_in,
                              void* d_out, int out_size, void* d_ws, size_t ws_size,
                              hipStream_t stream) {
  const float* x       = (const float*)d_in[0];
  const float* cd_w    = (const float*)d_in[1];
  const float* cd_b    = (const float*)d_in[2];
  const float* beta_w  = (const float*)d_in[3];
  const float* gamma_w = (const float*)d_in[4];
  const float* w_stack = (const float*)d_in[5];
  const float* w_nss   = (const float*)d_in[6];
  const float* b_nss   = (const float*)d_in[7];
  const float* w_nst   = (const float*)d_in[8];
  const float* w_st    = (const float*)d_in[9];
  const float* b_st    = (const float*)d_in[10];
  const float* w_sn    = (const float*)d_in[11];
  const float* b_sn    = (const float*)d_in[12];
  float* out = (float*)d_out;
  float* ws  = (float*)d_ws;

  // fit block-partial table inside the provided workspace (deterministic:
  // ws_size is fixed for the whole run)
  int nblk = NBLK;
  const size_t need = (size_t)NCOMP * sizeof(float);
  if (ws_size < (size_t)nblk * need) {
    nblk = (int)(ws_size / need);
    if (nblk < 1) nblk = 1;
  }

  ss1_stream_k1<<<nblk, NTHR, 0, stream>>>(x, cd_w, cd_b, beta_w, gamma_w, ws);
  ss1_finish_k2<<<1, 32, 0, stream>>>(x, cd_w, cd_b, beta_w, gamma_w,
                                      w_stack, w_nss, b_nss, w_nst,
                                      w_st, b_st, w_sn, b_sn, ws, nblk, out);
}